// HashEncoding_32332513804722
// MI455X (gfx1250) — compile-verified
//
#include <hip/hip_runtime.h>
#include <math.h>

#define NUM_LEVELS 16
#define LOG2_TABLE 19
#define TABLE_MASK ((1u << LOG2_TABLE) - 1u)

typedef float v2f __attribute__((ext_vector_type(2)));

struct Scales { float s[NUM_LEVELS]; };

__global__ __launch_bounds__(256) void hash_encoding_kernel(
    const float* __restrict__ in,   // [N, 3] float32
    const v2f*   __restrict__ ht,   // [16 * 2^19] float2 features
    v2f*         __restrict__ out,  // [N * 16] float2 (== [N, 32] float)
    Scales sc, int npoints)
{
    // Broadcast per-level scalings (host-computed, numpy-exact) through LDS.
    __shared__ float ls[NUM_LEVELS];
    if (threadIdx.x == 0) {
#pragma unroll
        for (int i = 0; i < NUM_LEVELS; ++i) ls[i] = sc.s[i];
    }
    __syncthreads();

    const int tid = blockIdx.x * blockDim.x + threadIdx.x;
    const int pt  = tid >> 4;     // point index
    const int lvl = tid & 15;     // level index
    if (pt >= npoints) return;

    // Streamed input: non-temporal (read exactly once per 16 lanes of one wave).
    const float x = __builtin_nontemporal_load(in + 3 * pt + 0);
    const float y = __builtin_nontemporal_load(in + 3 * pt + 1);
    const float z = __builtin_nontemporal_load(in + 3 * pt + 2);

    const float s  = ls[lvl];
    const float sx = x * s, sy = y * s, sz = z * s;
    const float fx = floorf(sx), fy = floorf(sy), fz = floorf(sz);
    const float ox = sx - fx,    oy = sy - fy,    oz = sz - fz;

    const unsigned fxi = (unsigned)fx;
    const unsigned fyi = (unsigned)fy;
    const unsigned fzi = (unsigned)fz;
    const unsigned cxi = (unsigned)ceilf(sx);
    const unsigned cyi = (unsigned)ceilf(sy);
    const unsigned czi = (unsigned)ceilf(sz);

    // h = x*1 ^ y*2654435761 ^ z*805459861 ; share the 6 partial products.
    const unsigned P2 = 2654435761u, P3 = 805459861u;
    const unsigned yf = fyi * P2, yc = cyi * P2;
    const unsigned zf = fzi * P3, zc = czi * P3;
    const unsigned lofs = ((unsigned)lvl) << LOG2_TABLE;

#define HGET(bx, by, bz) \
    ht[((((bx) ? cxi : fxi) ^ ((by) ? yc : yf) ^ ((bz) ? zc : zf)) & TABLE_MASK) + lofs]

    // CORNERS order from the reference:
    const v2f f0 = HGET(1, 1, 1);
    const v2f f1 = HGET(1, 0, 1);
    const v2f f2 = HGET(0, 0, 1);
    const v2f f3 = HGET(0, 1, 1);
    const v2f f4 = HGET(1, 1, 0);
    const v2f f5 = HGET(1, 0, 0);
    const v2f f6 = HGET(0, 0, 0);
    const v2f f7 = HGET(0, 1, 0);
#undef HGET

    const float wx1 = ox, wx0 = 1.0f - ox;
    const float wy1 = oy, wy0 = 1.0f - oy;
    const float wz1 = oz, wz0 = 1.0f - oz;

    // Trilinear lerp tree, exactly as the reference pairs corners.
    const v2f f03 = f0 * wx1 + f3 * wx0;
    const v2f f12 = f1 * wx1 + f2 * wx0;
    const v2f f47 = f4 * wx1 + f7 * wx0;
    const v2f f56 = f5 * wx1 + f6 * wx0;

    const v2f f0312 = f03 * wy1 + f12 * wy0;
    const v2f f4756 = f47 * wy1 + f56 * wy0;

    const v2f enc = f0312 * wz1 + f4756 * wz0;

    // Output is written once and never re-read: non-temporal store so the
    // 128 MB result stream does not evict the 64 MB hash table from L2.
    __builtin_nontemporal_store(enc, out + tid);
}

extern "C" void kernel_launch(void* const* d_in, const int* in_sizes, int n_in,
                              void* d_out, int out_size, void* d_ws, size_t ws_size,
                              hipStream_t stream) {
    const float* in = (const float*)d_in[0];
    const v2f*   ht = (const v2f*)d_in[1];
    v2f*         out = (v2f*)d_out;

    const int npoints = in_sizes[0] / 3;

    // Replicate numpy bit-exactly with host libm (glibc log/exp/pow — the same
    // routines numpy's float64 ufuncs call): floor(16 * growth^l) sits exactly
    // on integer boundaries at l = 5, 10, 15, so we must not re-derive it with
    // different math.
    Scales sc;
    const double growth = exp((log(1024.0) - log(16.0)) / 15.0);
    for (int l = 0; l < NUM_LEVELS; ++l)
        sc.s[l] = (float)floor(16.0 * pow(growth, (double)l));

    const long long total = (long long)npoints * NUM_LEVELS;
    const int threads = 256;
    const int blocks = (int)((total + threads - 1) / threads);

    hash_encoding_kernel<<<blocks, threads, 0, stream>>>(in, ht, out, sc, npoints);
}